// QCONV1_21655225106831
// MI455X (gfx1250) — compile-verified
//
#include <hip/hip_runtime.h>
#include <hip/hip_bf16.h>

typedef __attribute__((ext_vector_type(2))) float v2f;
typedef __attribute__((ext_vector_type(8))) float v8f;

// ---------------------------------------------------------------------------
// Kernel 1: per-patch 4-qubit statevector circuit (2 layers RY + CNOT ring),
// one thread per patch, 16 amplitudes in registers.
// Qubit q <-> bit position (3-q) of the flattened state index.
// ---------------------------------------------------------------------------

template <int BP>
__device__ __forceinline__ void apply_ry(float (&s)[16], float c, float sn) {
#pragma unroll
    for (int idx = 0; idx < 16; ++idx) {
        if (((idx >> BP) & 1) == 0) {
            const int h = idx | (1 << BP);
            const float lo = s[idx], hi = s[h];
            s[idx] = c * lo - sn * hi;
            s[h]   = sn * lo + c * hi;
        }
    }
}

template <int CP, int TP>
__device__ __forceinline__ void apply_cnot(float (&s)[16]) {
#pragma unroll
    for (int idx = 0; idx < 16; ++idx) {
        if (((idx >> CP) & 1) == 1 && ((idx >> TP) & 1) == 0) {
            const int h = idx | (1 << TP);
            const float tmp = s[idx];
            s[idx] = s[h];
            s[h] = tmp;
        }
    }
}

__global__ __launch_bounds__(128) void qconv_circuit_kernel(
    const float* __restrict__ X,   // (8,1,24,24)
    float* __restrict__ v)         // (1152*4) feature vector
{
    const int p = blockIdx.x * blockDim.x + threadIdx.x;   // patch id, 0..1151
    if (p >= 1152) return;

    const int b   = p / 144;
    const int rem = p - b * 144;
    const int i   = rem / 12;
    const int j   = rem - i * 12;

    // 2x2 patch -> angles (di-major, dj-minor): q0=(0,0) q1=(0,1) q2=(1,0) q3=(1,1)
    const float* xb = X + b * 576 + (2 * i) * 24 + 2 * j;
    const float t0 = xb[0],  t1 = xb[1];
    const float t2 = xb[24], t3 = xb[25];

    // Same angles are reused in both layers.
    float c0, s0, c1, s1, c2, s2, c3, s3;
    sincosf(0.5f * t0, &s0, &c0);
    sincosf(0.5f * t1, &s1, &c1);
    sincosf(0.5f * t2, &s2, &c2);
    sincosf(0.5f * t3, &s3, &c3);

    float st[16];
#pragma unroll
    for (int k = 0; k < 16; ++k) st[k] = 0.0f;
    st[0] = 1.0f;

#pragma unroll
    for (int layer = 0; layer < 2; ++layer) {
        apply_ry<3>(st, c0, s0);   // RY on qubit 0 (bit 3)
        apply_ry<2>(st, c1, s1);   // qubit 1
        apply_ry<1>(st, c2, s2);   // qubit 2
        apply_ry<0>(st, c3, s3);   // qubit 3
        apply_cnot<3, 2>(st);      // CNOT(0,1)
        apply_cnot<2, 1>(st);      // CNOT(1,2)
        apply_cnot<1, 0>(st);      // CNOT(2,3)
        apply_cnot<0, 3>(st);      // CNOT(3,0)
    }

    float z0 = 0.f, z1 = 0.f, z2 = 0.f, z3 = 0.f;
#pragma unroll
    for (int idx = 0; idx < 16; ++idx) {
        const float pr = st[idx] * st[idx];
        z0 += ((idx >> 3) & 1) ? -pr : pr;
        z1 += ((idx >> 2) & 1) ? -pr : pr;
        z2 += ((idx >> 1) & 1) ? -pr : pr;
        z3 += ( idx       & 1) ? -pr : pr;
    }

    reinterpret_cast<float4*>(v)[p] = make_float4(z0, z1, z2, z3);
}

// ---------------------------------------------------------------------------
// Kernel 2: out[0..9] = W(10x4608) @ v(4608) + bias, via V_WMMA_F32_16X16X4_F32.
// One workgroup, 32 waves; wave w reduces K-slice [w*144, w*144+144).
// A layout (16x4 f32): lanes 0-15 carry K=0/1 in the 2 VGPRs, lanes 16-31 K=2/3.
// B (4x16) is v broadcast across all N columns; only column N=0 is consumed.
// Rows 10..15 of A use a CLAMPED row pointer (valid memory, garbage values):
// D rows depend only on their own A row, and rows 10..15 of D are never read,
// so no masking is needed -> branchless loop body (4x b64 load + 2x wmma).
// C/D layout: VGPR r -> M=r (lanes 0-15) / M=8+r (lanes 16-31), N=lane%16.
// Deterministic LDS reduction, serial over the 32 waves (no float atomics).
// ---------------------------------------------------------------------------

__global__ __launch_bounds__(1024) void qconv_cls_wmma_kernel(
    const float* __restrict__ W,     // (10, 4608) row-major
    const float* __restrict__ bias,  // (10,)
    const float* __restrict__ v,     // (4608,)
    float* __restrict__ out)         // (10,)
{
    __shared__ float partial[32][16];

    const int tid  = threadIdx.x;
    const int wave = tid >> 5;
    const int lane = tid & 31;
    const int m    = lane & 15;               // output row this lane feeds
    const int koff = (lane >> 4) << 1;        // 0 for lanes 0-15, 2 for lanes 16-31
    const int mrow = (m < 10) ? m : 9;        // clamp: rows 10..15 of D are discarded

    const int kbase = wave * 144;
    const float2* __restrict__ pW = reinterpret_cast<const float2*>(W + mrow * 4608 + kbase + koff);
    const float2* __restrict__ pV = reinterpret_cast<const float2*>(v + kbase + koff);

    v8f acc0 = {};
    v8f acc1 = {};

    // 144 K-values per wave = 18 iterations of 2 independent WMMA chains.
#pragma unroll 3
    for (int it = 0; it < 18; ++it) {
        // float2 index: K chunk [8*it, 8*it+4) -> elements {4*it + 0/1} of the
        // lane's half; chunk [8*it+4, 8*it+8) -> {4*it + 2/3}.  koff picks the
        // lane-half (K%4 in {0,1} vs {2,3}) per the 16x4 f32 A-layout.
        const float2 w0 = pW[4 * it];
        const float2 b0 = pV[4 * it];
        const float2 w1 = pW[4 * it + 2];
        const float2 b1 = pV[4 * it + 2];

        v2f a0; a0.x = w0.x; a0.y = w0.y;
        v2f v0; v0.x = b0.x; v0.y = b0.y;
        acc0 = __builtin_amdgcn_wmma_f32_16x16x4_f32(
            false, a0, false, v0, (short)0, acc0, false, false);

        v2f a1; a1.x = w1.x; a1.y = w1.y;
        v2f v1; v1.x = b1.x; v1.y = b1.y;
        acc1 = __builtin_amdgcn_wmma_f32_16x16x4_f32(
            false, a1, false, v1, (short)0, acc1, false, false);
    }

    // Extract column N=0 of D: lane 0 holds M=0..7 in acc[0..7], lane 16 holds M=8..15.
    if (lane == 0) {
#pragma unroll
        for (int r = 0; r < 8; ++r) partial[wave][r] = acc0[r] + acc1[r];
    } else if (lane == 16) {
#pragma unroll
        for (int r = 0; r < 8; ++r) partial[wave][8 + r] = acc0[r] + acc1[r];
    }
    __syncthreads();

    if (tid < 10) {
        float s = bias[tid];
#pragma unroll
        for (int w = 0; w < 32; ++w) s += partial[w][tid];
        out[tid] = s;
    }
}

// ---------------------------------------------------------------------------

extern "C" void kernel_launch(void* const* d_in, const int* in_sizes, int n_in,
                              void* d_out, int out_size, void* d_ws, size_t ws_size,
                              hipStream_t stream) {
    const float* X    = (const float*)d_in[0];   // (8,1,24,24)  = 4608 f32
    const float* Wc   = (const float*)d_in[1];   // (10,4608)    = 46080 f32
    const float* bias = (const float*)d_in[2];   // (10,)
    float* v   = (float*)d_ws;                   // 4608 f32 scratch (18.4 KB)
    float* out = (float*)d_out;                  // (10,)

    qconv_circuit_kernel<<<dim3(9), dim3(128), 0, stream>>>(X, v);
    qconv_cls_wmma_kernel<<<dim3(1), dim3(1024), 0, stream>>>(Wc, bias, v, out);
}